// ProbAttention_62723702391036
// MI455X (gfx1250) — compile-verified
//
#include <hip/hip_runtime.h>

#define BB   4
#define LL   2048
#define HH   8
#define EE   64
#define SK   40     // sample_k = 5*ceil(ln(2049)) = 40
#define NTOP 40     // n_top, same formula
#define RS   (HH*EE)  // row stride in (B,L,H,E) layout = 512 floats

typedef __attribute__((ext_vector_type(2))) float v2f;
typedef __attribute__((ext_vector_type(8))) float v8f;

// ---------------------------------------------------------------------------
// Kernel 1: sparsity[b,h,l] = max_s(q.k_sample) - mean_s(q.k_sample)
// ---------------------------------------------------------------------------
__global__ void sparsity_kernel(const float* __restrict__ q,
                                const float* __restrict__ k,
                                const int*   __restrict__ idxs,
                                float* __restrict__ sp) {
    int gid = blockIdx.x * blockDim.x + threadIdx.x;   // (b*H + h)*L + l
    if (gid >= BB*HH*LL) return;
    int l = gid % LL;
    int h = (gid / LL) % HH;
    int b = gid / (LL*HH);

    const float4* q4 = (const float4*)(q + ((size_t)(b*LL + l)*HH + h)*EE);
    float4 qv[16];
#pragma unroll
    for (int i = 0; i < 16; ++i) qv[i] = q4[i];

    float mx = -1e30f, sum = 0.0f;
    for (int s = 0; s < SK; ++s) {
        int ki = idxs[l*SK + s];
        const float4* k4 = (const float4*)(k + ((size_t)(b*LL + ki)*HH + h)*EE);
        float d = 0.0f;
#pragma unroll
        for (int i = 0; i < 16; ++i) {
            float4 kv = k4[i];
            d += qv[i].x*kv.x + qv[i].y*kv.y + qv[i].z*kv.z + qv[i].w*kv.w;
        }
        mx = fmaxf(mx, d);
        sum += d;
    }
    sp[gid] = mx - sum * (1.0f/(float)SK);
}

// ---------------------------------------------------------------------------
// Kernel 2: per (b,h) top-40 indices (jax tie-break: smaller index on ties)
// Mutates sp (workspace copy regenerated by kernel 1 each launch).
// ---------------------------------------------------------------------------
__global__ void topk_kernel(float* __restrict__ sp, int* __restrict__ top) {
    int bh = blockIdx.x;               // 0..31
    float* s = sp + (size_t)bh * LL;
    __shared__ float sv[256];
    __shared__ int   si[256];
    int t = threadIdx.x;

    for (int n = 0; n < NTOP; ++n) {
        float bv = -1e38f; int bi = LL;
        for (int i = t; i < LL; i += 256) {
            float v = s[i];
            if (v > bv || (v == bv && i < bi)) { bv = v; bi = i; }
        }
        sv[t] = bv; si[t] = bi;
        __syncthreads();
        for (int off = 128; off > 0; off >>= 1) {
            if (t < off) {
                float ov = sv[t+off]; int oi = si[t+off];
                if (ov > sv[t] || (ov == sv[t] && oi < si[t])) { sv[t] = ov; si[t] = oi; }
            }
            __syncthreads();
        }
        int w = si[0];
        if (t == 0) { top[bh*NTOP + n] = w; s[w] = -1e38f; }
        __syncthreads();
    }
}

// ---------------------------------------------------------------------------
// Kernel 3: flash-style causal attention for the top queries.
// One wave32 per (b, h, 16-query tile). Uses V_WMMA_F32_16X16X4_F32.
// ---------------------------------------------------------------------------
__global__ void __launch_bounds__(32)
attn_kernel(const float* __restrict__ q,
            const float* __restrict__ kmat,
            const float* __restrict__ vmat,
            const int*   __restrict__ top,
            float* __restrict__ upd) {
    __shared__ float pbuf[16*17];

    int blk  = blockIdx.x;          // bh*3 + tile
    int tile = blk % 3;
    int bh   = blk / 3;
    int h = bh % HH, b = bh / HH;
    int lane = threadIdx.x;         // 0..31
    int half = lane >> 4;           // 0/1
    int n16  = lane & 15;

    const float* kbh = kmat + (size_t)b*LL*RS + h*EE;   // row(key) = kbh + key*RS
    const float* vbh = vmat + (size_t)b*LL*RS + h*EE;

    // causal positions of this lane's 8 C-layout rows (m = i + half*8)
    int qpos[8];
#pragma unroll
    for (int i = 0; i < 8; ++i) {
        int mg = tile*16 + i + half*8;
        qpos[i] = (mg < NTOP) ? top[bh*NTOP + mg] : -1;   // -1: padded row, fully masked
    }

    // Q tile in A-layout (16x4 per chunk, 16 chunks over E=64):
    // lane n16 = row m, lanes<16 get K pair {0,1}, lanes>=16 get {2,3}
    int mgA  = tile*16 + n16;
    int qrow = (mgA < NTOP) ? top[bh*NTOP + mgA] : 0;
    const float* qp = q + ((size_t)(b*LL + qrow)*HH + h)*EE;
    v2f aQ[16];
#pragma unroll
    for (int c = 0; c < 16; ++c) {
        int e = c*4 + half*2;
        aQ[c].x = qp[e];
        aQ[c].y = qp[e+1];
    }

    v8f acc[4];
#pragma unroll
    for (int t = 0; t < 4; ++t)
#pragma unroll
        for (int i = 0; i < 8; ++i) acc[t][i] = 0.0f;
    float runmax[8], runsum[8];
#pragma unroll
    for (int i = 0; i < 8; ++i) { runmax[i] = -1e30f; runsum[i] = 0.0f; }

    for (int j = 0; j < LL/16; ++j) {
        int kb = j*16;

        // ---- S tile = Qtile(16xE) * K_block^T(Ex16), 16 WMMAs of K=4 ----
        v8f S;
#pragma unroll
        for (int i = 0; i < 8; ++i) S[i] = 0.0f;
        const float* krow = kbh + (size_t)(kb + n16)*RS;   // B: lane n -> key column n
#pragma unroll
        for (int c = 0; c < 16; ++c) {
            int e = c*4 + half*2;                          // B v0: rows K0/K2, v1: K1/K3
            v2f bK; bK.x = krow[e]; bK.y = krow[e+1];
            S = __builtin_amdgcn_wmma_f32_16x16x4_f32(false, aQ[c], false, bK,
                                                      (short)0, S, false, false);
        }

        // ---- scale + causal mask; C layout: (m = i+half*8, n = n16) ----
        int kpos = kb + n16;
        float sv[8];
#pragma unroll
        for (int i = 0; i < 8; ++i) {
            float x = S[i] * 0.125f;                       // 1/sqrt(64)
            if (kpos > qpos[i]) x = -1e30f;
            sv[i] = x;
        }

        // ---- online softmax: row reductions across the 16-lane half ----
#pragma unroll
        for (int i = 0; i < 8; ++i) {
            float m = sv[i];
#pragma unroll
            for (int off = 1; off < 16; off <<= 1) m = fmaxf(m, __shfl_xor(m, off, 16));
            float nm  = fmaxf(runmax[i], m);
            float fac = __expf(runmax[i] - nm);
            runmax[i] = nm;
            float p = __expf(sv[i] - nm);
            sv[i] = p;
            float rs = p;
#pragma unroll
            for (int off = 1; off < 16; off <<= 1) rs += __shfl_xor(rs, off, 16);
            runsum[i] = runsum[i]*fac + rs;
#pragma unroll
            for (int t = 0; t < 4; ++t) acc[t][i] *= fac;
        }

        // ---- transpose P: C-layout -> A-layout via LDS ----
        __syncthreads();
#pragma unroll
        for (int i = 0; i < 8; ++i)
            pbuf[(i + half*8)*17 + n16] = sv[i];
        __syncthreads();
        v2f aP[4];
#pragma unroll
        for (int c = 0; c < 4; ++c) {
            int kk = c*4 + half*2;
            aP[c].x = pbuf[n16*17 + kk];
            aP[c].y = pbuf[n16*17 + kk + 1];
        }

        // ---- acc(16xE) += P(16x16) * V_block(16xE), 16 WMMAs ----
#pragma unroll
        for (int c = 0; c < 4; ++c) {
            int key0 = kb + c*4 + half*2;
            const float* v0 = vbh + (size_t)key0*RS;
            const float* v1 = vbh + (size_t)(key0+1)*RS;
#pragma unroll
            for (int t = 0; t < 4; ++t) {
                v2f bV; bV.x = v0[t*16 + n16]; bV.y = v1[t*16 + n16];
                acc[t] = __builtin_amdgcn_wmma_f32_16x16x4_f32(false, aP[c], false, bV,
                                                               (short)0, acc[t], false, false);
            }
        }
        __syncthreads();   // pbuf reused next iteration
    }

    // ---- epilogue: normalize, store valid rows to updates workspace ----
#pragma unroll
    for (int i = 0; i < 8; ++i) {
        int mg = tile*16 + i + half*8;
        if (mg < NTOP) {
            float inv = 1.0f / runsum[i];
#pragma unroll
            for (int t = 0; t < 4; ++t)
                upd[((size_t)bh*NTOP + mg)*EE + t*16 + n16] = acc[t][i] * inv;
        }
    }
}

// ---------------------------------------------------------------------------
// Kernel 4: context = cumsum(V over L); overwrite top rows; out is (B,L,H,E)
// ---------------------------------------------------------------------------
__global__ void cumsum_scatter_kernel(const float* __restrict__ vmat,
                                      const int*   __restrict__ top,
                                      const float* __restrict__ upd,
                                      float* __restrict__ out) {
    int bh = blockIdx.x;
    int h = bh % HH, b = bh / HH;
    int e = threadIdx.x;                                  // 0..63
    const float* vp = vmat + (size_t)b*LL*RS + h*EE + e;
    float*       op = out  + (size_t)b*LL*RS + h*EE + e;
    float run = 0.0f;
    for (int l = 0; l < LL; ++l) {
        run += vp[(size_t)l*RS];
        op[(size_t)l*RS] = run;
    }
    for (int n = 0; n < NTOP; ++n) {
        int pos = top[bh*NTOP + n];
        op[(size_t)pos*RS] = upd[((size_t)bh*NTOP + n)*EE + e];
    }
}

// ---------------------------------------------------------------------------
extern "C" void kernel_launch(void* const* d_in, const int* in_sizes, int n_in,
                              void* d_out, int out_size, void* d_ws, size_t ws_size,
                              hipStream_t stream) {
    const float* q    = (const float*)d_in[0];
    const float* k    = (const float*)d_in[1];
    const float* v    = (const float*)d_in[2];
    const int*   idxs = (const int*)d_in[3];
    float* out = (float*)d_out;

    char* ws = (char*)d_ws;
    float* sp  = (float*)ws;                                         // B*H*L f32
    size_t off = (size_t)BB*HH*LL*sizeof(float);
    int*   top = (int*)(ws + off);                                   // B*H*40 i32
    off += ((size_t)BB*HH*NTOP*sizeof(int) + 255) & ~(size_t)255;
    float* upd = (float*)(ws + off);                                 // B*H*40*64 f32

    sparsity_kernel<<<(BB*HH*LL + 255)/256, 256, 0, stream>>>(q, k, idxs, sp);
    topk_kernel<<<BB*HH, 256, 0, stream>>>(sp, top);
    attn_kernel<<<BB*HH*3, 32, 0, stream>>>(q, k, v, top, upd);
    cumsum_scatter_kernel<<<BB*HH, EE, 0, stream>>>(v, top, upd, out);
}